// SessionGraph_51367808860292
// MI455X (gfx1250) — compile-verified
//
#include <hip/hip_runtime.h>
#include <hip/hip_bf16.h>

typedef __attribute__((ext_vector_type(2))) float v2f;
typedef __attribute__((ext_vector_type(8))) float v8f;
typedef __attribute__((ext_vector_type(4))) unsigned int v4u;
typedef __attribute__((ext_vector_type(8))) int v8i;
typedef __attribute__((ext_vector_type(4))) int v4i;

#define BDIM 512
#define NDIM 256
#define HDIM 128
#define ALPHA_C 0.8f
#define BETA_C  0.1f

// ---------------------------------------------------------------------------
// Kernel 1: row sums of folded adjacency (+1 self loop) -> d_inv_sqrt [B*N]
// ---------------------------------------------------------------------------
__global__ __launch_bounds__(256) void sgc_rowsum(const float* __restrict__ A,
                                                  float* __restrict__ dinv) {
  const int row  = (blockIdx.x * blockDim.x + threadIdx.x) >> 5;   // b*N + i
  const int lane = threadIdx.x & 31;
  const float* r = A + (size_t)row * (2 * NDIM);
  float s = 0.f;
#pragma unroll
  for (int t = 0; t < NDIM; t += 32)
    s += r[t + lane] + r[NDIM + t + lane];
#pragma unroll
  for (int off = 16; off > 0; off >>= 1)
    s += __shfl_xor(s, off, 32);
  s += 1.0f;                                   // self loop contribution
  float d = (s > 0.f) ? (1.0f / sqrtf(s)) : 0.f;
  if (lane == 0) dinv[row] = d;
}

// ---------------------------------------------------------------------------
// Kernel 2: A_mix = alpha * D^-1/2 (A_in+A_out+I) D^-1/2 + beta*(1-norm)*I
// ---------------------------------------------------------------------------
__global__ __launch_bounds__(256) void sgc_build_amix(const float* __restrict__ A,
                                                      const float* __restrict__ dinv,
                                                      float* __restrict__ Amix) {
  const size_t tid = (size_t)blockIdx.x * blockDim.x + threadIdx.x;
  const size_t row = tid / (NDIM / 4);                  // b*N + i
  const int    j0  = (int)(tid % (NDIM / 4)) * 4;
  const int    i   = (int)(row % NDIM);
  const size_t b   = row / NDIM;

  const float* ar = A + row * (2 * NDIM);
  const float  di = dinv[row];
  float4 a0 = *(const float4*)(ar + j0);
  float4 a1 = *(const float4*)(ar + NDIM + j0);
  float4 dj = *(const float4*)(dinv + b * NDIM + j0);

  const float av[4] = {a0.x + a1.x, a0.y + a1.y, a0.z + a1.z, a0.w + a1.w};
  const float dv[4] = {dj.x, dj.y, dj.z, dj.w};
  float o[4];
#pragma unroll
  for (int c = 0; c < 4; ++c) {
    const bool diag = (j0 + c) == i;
    const float adj = av[c] + (diag ? 1.f : 0.f);
    const float nrm = adj * di * dv[c];
    o[c] = ALPHA_C * nrm + (diag ? BETA_C * (1.f - nrm) : 0.f);
  }
  *(float4*)(Amix + row * NDIM + j0) = make_float4(o[0], o[1], o[2], o[3]);
}

// ---------------------------------------------------------------------------
// Kernel 3: embedding gather. One wave per (b,n); float4 per lane (32*4=128).
// ---------------------------------------------------------------------------
__global__ __launch_bounds__(256) void sgc_gather(const int* __restrict__ idx,
                                                  const float* __restrict__ emb,
                                                  float* __restrict__ out) {
  const int row  = (blockIdx.x * blockDim.x + threadIdx.x) >> 5;
  const int lane = threadIdx.x & 31;
  const int node = idx[row];
  const float4* src = (const float4*)(emb + (size_t)node * HDIM);
  float4*       dst = (float4*)(out + (size_t)row * HDIM);
  dst[lane] = src[lane];
}

// ---------------------------------------------------------------------------
// Kernel 4: batched fp32 GEMM Y[b] = Amix[b] (256x256) * X[b] (256x128)
// with V_WMMA_F32_16X16X4_F32.
//  * grid = B*2 ; block (8 waves) computes a 128x128 output half-batch.
//  * A K-chunk (128 rows x 32 cols) is DMA'd by the Tensor Data Mover
//    (tensor_load_to_lds) into LDS with a padded row stride of 36 floats
//    (pad_interval = 32 DW, pad_amount = 4 DW) -> conflict-free ds_load_b64
//    A-fragments. Wave 0 issues the TDM, waits on TENSORcnt, block barriers.
//  * X K-chunk is staged manually, K-pair interleaved:
//      lds[((k>>1)*144 + n)*2 + (k&1)]
//    so each B-fragment (K=k..k+1 at col n) is one contiguous ds_load_b64
//    straight into an even-aligned VGPR pair (no v_mov shuffling), and the
//    two lane-halves of the load land on disjoint bank groups (2*144 % 64 = 32).
// ---------------------------------------------------------------------------
__global__ __launch_bounds__(256) void sgc_spmm_wmma(const float* __restrict__ Amix,
                                                     const float* __restrict__ X,
                                                     float* __restrict__ Y) {
  constexpr int KC   = 32;          // K chunk
  constexpr int LDA  = KC + 4;      // padded A row stride (floats), via TDM pad
  constexpr int LDNX = 144;         // X pair-stride: 2*144 % 64 == 32
  __shared__ float ldsA[128 * LDA];           // 18 KB
  __shared__ float ldsX[(KC / 2) * LDNX * 2]; // 18 KB

  const int b      = blockIdx.x >> 1;
  const int m_base = (blockIdx.x & 1) * 128;
  const int wave   = threadIdx.x >> 5;
  const int lane   = threadIdx.x & 31;

  const float* __restrict__ M  = Amix + (size_t)b * NDIM * NDIM;
  const float* __restrict__ Xb = X    + (size_t)b * NDIM * HDIM;
  float*       __restrict__ Yb = Y    + (size_t)b * NDIM * HDIM;

  v8f acc[8];
#pragma unroll
  for (int t = 0; t < 8; ++t)
#pragma unroll
    for (int e = 0; e < 8; ++e) acc[t][e] = 0.f;

  const int lm = lane & 15;
  const int kh = (lane >> 4) << 1;              // 0 or 2: lane-half K offset
  const unsigned lds_a_base = (unsigned)(uintptr_t)(void*)ldsA;

  for (int ic = 0; ic < NDIM / KC; ++ic) {
    const int kc = ic * KC;

    // --- wave 0: issue TDM load of A tile [m_base..+127] x [kc..kc+31] ---
    if (threadIdx.x < 32) {
      const unsigned long long ga =
          (unsigned long long)(uintptr_t)(M + (size_t)m_base * NDIM + kc);
      v4u g0;
      g0[0] = 1u;                                           // count=1, user desc
      g0[1] = lds_a_base;                                   // lds_addr
      g0[2] = (unsigned)(ga & 0xFFFFFFFFu);                 // global_addr lo
      g0[3] = (unsigned)((ga >> 32) & 0x1FFFFFFu) | (2u << 30); // addr hi | type=2
      v8i g1;
      g1[0] = (int)((2u << 16)      // data_size = 4B
                  | (1u << 20)      // pad_enable
                  | (4u << 22)      // pad_interval: 32 DWORDs
                  | (3u << 25));    // pad_amount: 4 DWORDs
      g1[1] = (int)(256u << 16);    // tensor_dim0 = 256 (lo16 @ bits[63:48])
      g1[2] = 0;                    // tensor_dim0 hi, tensor_dim1 lo16 = 0
      g1[3] = (int)(1u | (32u << 16)); // tensor_dim1 = 65536 ; tile_dim0 = 32
      g1[4] = 128;                  // tile_dim1 = 128 ; tile_dim2 = 0
      g1[5] = 256;                  // tensor_dim0_stride = 256
      g1[6] = 0;
      g1[7] = 0;
      v4i g2 = {0, 0, 0, 0};
      v4i g3 = {0, 0, 0, 0};
      v8i g4 = {0, 0, 0, 0, 0, 0, 0, 0};   // unused extended group (clang-23 form)
      __builtin_amdgcn_tensor_load_to_lds(g0, g1, g2, g3, g4, 0);
    }

    // --- all threads: stage X chunk, K-pair interleaved (overlaps the TDM) ---
#pragma unroll
    for (int t = 0; t < 2; ++t) {
      const int e = threadIdx.x + t * 256;      // 0..511 tasks
      const int p = e >> 5;                     // K-pair index 0..15
      const int c = (e & 31) << 2;              // column quad 0..124
      float4 x0 = *(const float4*)(Xb + (size_t)(kc + 2 * p) * HDIM + c);
      float4 x1 = *(const float4*)(Xb + (size_t)(kc + 2 * p + 1) * HDIM + c);
      float* d = &ldsX[(p * LDNX + c) * 2];
      ((float4*)d)[0] = make_float4(x0.x, x1.x, x0.y, x1.y);
      ((float4*)d)[1] = make_float4(x0.z, x1.z, x0.w, x1.w);
    }

    if (threadIdx.x < 32) __builtin_amdgcn_s_wait_tensorcnt(0);
    __syncthreads();

    // --- compute: per K-step 1 A ds_load_b64 + 8 B ds_load_b64 + 8 WMMA ---
    const float* arow = &ldsA[(wave * 16 + lm) * LDA + kh];
#pragma unroll
    for (int k = 0; k < KC; k += 4) {
      v2f afrag = *(const v2f*)(arow + k);           // A: K = k+kh, k+kh+1
      const int pk = (k >> 1) + (lane >> 4);         // (k+kh)/2
#pragma unroll
      for (int nt = 0; nt < 8; ++nt) {
        v2f bfrag = *(const v2f*)&ldsX[(pk * LDNX + nt * 16 + lm) * 2];
        acc[nt] = __builtin_amdgcn_wmma_f32_16x16x4_f32(
            false, afrag, false, bfrag, (short)0, acc[nt], false, false);
      }
    }
    __syncthreads();
  }

  // C/D layout: lane<16 -> M = m0+r, N = lane ; lane>=16 -> M = m0+8+r
  const int m0 = m_base + wave * 16;
  const int mh = (lane >> 4) << 3;
#pragma unroll
  for (int nt = 0; nt < 8; ++nt) {
#pragma unroll
    for (int r = 0; r < 8; ++r) {
      Yb[(size_t)(m0 + r + mh) * HDIM + nt * 16 + lm] = acc[nt][r];
    }
  }
}

// ---------------------------------------------------------------------------
extern "C" void kernel_launch(void* const* d_in, const int* in_sizes, int n_in,
                              void* d_out, int out_size, void* d_ws, size_t ws_size,
                              hipStream_t stream) {
  (void)in_sizes; (void)n_in; (void)out_size; (void)ws_size;
  const int*   inputs = (const int*)d_in[0];
  const float* A      = (const float*)d_in[1];
  const float* emb    = (const float*)d_in[2];
  float*       out    = (float*)d_out;

  float* ws   = (float*)d_ws;
  float* dinv = ws;                                        //  B*N      floats
  float* Amix = ws + (size_t)BDIM * NDIM;                  //  B*N*N    floats
  float* H0   = Amix + (size_t)BDIM * NDIM * NDIM;         //  B*N*H    floats
  float* H1   = H0 + (size_t)BDIM * NDIM * HDIM;           //  B*N*H    floats

  const unsigned rows = BDIM * NDIM;                       // 131072

  sgc_rowsum    <<<rows / 8, 256, 0, stream>>>(A, dinv);
  sgc_build_amix<<<rows * (NDIM / 4) / 256, 256, 0, stream>>>(A, dinv, Amix);
  sgc_gather    <<<rows / 8, 256, 0, stream>>>(inputs, emb, H0);
  sgc_spmm_wmma <<<BDIM * 2, 256, 0, stream>>>(Amix, H0, H1);
  sgc_spmm_wmma <<<BDIM * 2, 256, 0, stream>>>(Amix, H1, out);
}